// ChamferDistance_34789235097880
// MI455X (gfx1250) — compile-verified
//
#include <hip/hip_runtime.h>
#include <hip/hip_bf16.h>

typedef __attribute__((ext_vector_type(2))) float v2f;
typedef __attribute__((ext_vector_type(8))) float v8f;

// One wave computes min_j d(p_i, q_j) for 16 query points i, sweeping all Nb
// target points j in 16-wide tiles via V_WMMA_F32_16X16X4_F32.
//
// Augmented K=8 GEMM (zero padded past K=4):
//   A row  (query p):  [ px, py, pz, ||p||² ]  then [ 1, 0, 0, 0 ]
//   B col  (target q): [-2qx,-2qy,-2qz, 1   ]  then [ ||q||², 0, 0, 0 ]
//   A·B = ||p||² + ||q||² - 2 p·q
//
// A 16x4 f32 layout (ISA 7.12.2): lane = M (0..15 in each half),
//   lanes 0-15 hold K=0 (vgpr0), K=1 (vgpr1); lanes 16-31 hold K=2, K=3.
// B 4x16 f32 layout: lane = N; lanes 0-15 hold K=0/K=1, lanes 16-31 K=2/K=3.
// C/D 16x16 f32: lane = N (per half), vgpr v = M=v (lanes 0-15) / M=v+8.
__global__ __launch_bounds__(256) void chamfer_rowmin_wmma(
    const float* __restrict__ pa,   // [B, Na, 3] query points (rows of D)
    const float* __restrict__ pb,   // [B, Nb, 3] target points (cols of D)
    float* __restrict__ out,        // [B, Na]  out[i] = min_j d(i,j)
    int Na, int Nb)
{
  const int batch     = blockIdx.y;
  const int lane      = threadIdx.x & 31;
  const int wave      = threadIdx.x >> 5;
  const int rowBase   = blockIdx.x * 128 + wave * 16;   // 8 waves x 16 rows
  const int idxInHalf = lane & 15;
  const bool hi       = lane >= 16;                     // K-half selector

  // ---- Load this wave's 16 query points (both halves read same point) ----
  const float* ap = pa + ((size_t)batch * Na + (size_t)(rowBase + idxInHalf)) * 3;
  const float ax = ap[0], ay = ap[1], az = ap[2];
  const float asq = ax * ax + ay * ay + az * az;

  v2f a_lo, a_hi;
  a_lo.x = hi ? az   : ax;    // K=2 : K=0
  a_lo.y = hi ? asq  : ay;    // K=3 : K=1
  a_hi.x = hi ? 0.0f : 1.0f;  // K=6 : K=4
  a_hi.y = 0.0f;              // K=7 : K=5

  v8f minacc;
#pragma unroll
  for (int v = 0; v < 8; ++v) minacc[v] = 3.0e38f;

  const float* bbase = pb + (size_t)batch * Nb * 3;

  for (int col = 0; col < Nb; col += 16) {
    const float* bp = bbase + (size_t)(col + idxInHalf) * 3;
    // Prefetch next column tile (+16 points = +48 floats) into cache.
    __builtin_prefetch(bp + 48, 0, 1);

    const float bx = bp[0], by = bp[1], bz = bp[2];
    const float bsq = bx * bx + by * by + bz * bz;

    v2f b_lo, b_hi;
    b_lo.x = hi ? -2.0f * bz : -2.0f * bx;  // K=2 : K=0
    b_lo.y = hi ?  1.0f      : -2.0f * by;  // K=3 : K=1
    b_hi.x = hi ?  0.0f      :  bsq;        // K=6 : K=4
    b_hi.y = 0.0f;                          // K=7 : K=5

    v8f c = {};
    c = __builtin_amdgcn_wmma_f32_16x16x4_f32(false, a_lo, false, b_lo,
                                              (short)0, c, false, false);
    c = __builtin_amdgcn_wmma_f32_16x16x4_f32(false, a_hi, false, b_hi,
                                              (short)0, c, false, false);

#pragma unroll
    for (int v = 0; v < 8; ++v) minacc[v] = fminf(minacc[v], c[v]);
  }

  // ---- Min-reduce across the 16 lanes of each half (N dimension) ----
#pragma unroll
  for (int m = 8; m >= 1; m >>= 1) {
#pragma unroll
    for (int v = 0; v < 8; ++v) {
      const float o = __shfl_xor(minacc[v], m, 32);
      minacc[v] = fminf(minacc[v], o);
    }
  }

  // Lane 0 holds mins for M=0..7 (vgpr v -> row v); lane 16 for M=8..15.
  if (idxInHalf == 0) {
    float* o = out + (size_t)batch * Na + rowBase + (hi ? 8 : 0);
#pragma unroll
    for (int v = 0; v < 8; ++v) o[v] = minacc[v];
  }
}

extern "C" void kernel_launch(void* const* d_in, const int* in_sizes, int n_in,
                              void* d_out, int out_size, void* d_ws, size_t ws_size,
                              hipStream_t stream) {
  const float* xyz1 = (const float*)d_in[0];
  const float* xyz2 = (const float*)d_in[1];
  float* out = (float*)d_out;

  const int B = 4;
  const int N = in_sizes[0] / (B * 3);  // 8192
  const int M = in_sizes[1] / (B * 3);  // 8192

  dim3 block(256);
  dim3 grid1((unsigned)(N / 128), (unsigned)B);
  dim3 grid2((unsigned)(M / 128), (unsigned)B);

  // dist1[b][n] = min_m d(xyz1[n], xyz2[m])
  chamfer_rowmin_wmma<<<grid1, block, 0, stream>>>(xyz1, xyz2, out, N, M);
  // dist2[b][m] = min_n d(xyz2[m], xyz1[n])  (swap roles; no atomics needed)
  chamfer_rowmin_wmma<<<grid2, block, 0, stream>>>(xyz2, xyz1,
                                                   out + (size_t)B * N, M, N);
}